// AggregationLayer_29188597743703
// MI455X (gfx1250) — compile-verified
//
#include <hip/hip_runtime.h>
#include <math.h>

// ---------------- problem constants (match reference) ----------------
#define Bsz   8
#define HH    320
#define WW    320
#define HW    (HH * WW)          // 102400
#define IPI   16                 // instances per image
#define NN    (Bsz * IPI)        // 128 instances
#define NCH   9                  // q0..q3, s0..s2, z, ones(mask_size)

// work split for the WMMA reduction kernel
#define SPLITS          25                       // HW/SPLITS = 4096 px per block
#define PIX_PER_BLOCK   (HW / SPLITS)            // 4096
#define WAVES_PER_BLOCK 8                        // 256 threads, wave32
#define PIX_PER_WAVE    (PIX_PER_BLOCK / WAVES_PER_BLOCK)  // 512
#define STEPS           (PIX_PER_WAVE / 4)       // 128 WMMA k-steps of 4 px

// streaming kernel geometry: 25600 quads/instance, 256 quads/block
#define QPI             (HW / 4)                 // 25600
#define BLOCKS_PER_INST (QPI / 256)              // 100 (exact)

// workspace layout: [0, 2048) f32 sums, [2048, 2176) i32 class max
#define WS_WORDS  (NN * 16 + NN)                 // 2176

// flat output offsets (tuple concatenated in return order, all as float)
// (class_ids, masks, sample_ids, quat, sc, xy_masked, z_agg)
#define OFF_CLASS 0
#define OFF_MASKS 128
#define OFF_SIDS  (OFF_MASKS + (size_t)NN * HW)            // 13107328
#define OFF_QUAT  (OFF_SIDS + NN)                          // 13107456
#define OFF_SC    (OFF_QUAT + (size_t)NN * 4)              // 13107968
#define OFF_XY    (OFF_SC + (size_t)NN * 3)                // 13108352
#define OFF_Z     (OFF_XY + (size_t)NN * 2 * HW)           // 39322752

typedef __attribute__((ext_vector_type(2))) float v2f;
typedef __attribute__((ext_vector_type(4))) float v4f;
typedef __attribute__((ext_vector_type(8))) float v8f;
typedef __attribute__((ext_vector_type(4))) int   v4i;

// ---------------- kernel 0: zero the workspace ----------------
__global__ void zero_ws_kernel(int* __restrict__ ws, int n) {
    int i = blockIdx.x * blockDim.x + threadIdx.x;
    if (i < n) ws[i] = 0;
}

// ---------------- kernel 1: per-image masked sums via V_WMMA_F32_16X16X4_F32 --
// Per image b:  S[16 inst, 16 ch] += M_b[16, HW] @ F_b[HW, 16]
// Channels: 0..3 quaternion, 4..6 scales, 7 z, 8 ones (-> mask_size), 9..15 zero.
//
// A-matrix (16x4 f32, wave32):  lane L holds row M = L%16, K = 2*(L/16)+{0,1}
// B-matrix (4x16  f32, wave32): lane L holds col N = L%16, K = 2*(L/16)+{0,1}
// C/D (16x16 f32): VGPR v holds (M = v + 8*(L/16), N = L%16)
__global__ __launch_bounds__(256) void inst_sums_wmma_kernel(
    const float* __restrict__ masks,   // [N, H, W]
    const float* __restrict__ quat,    // [B, 4, H, W]
    const float* __restrict__ scales,  // [B, 3, H, W]
    const float* __restrict__ z,       // [B, H, W]
    float* __restrict__ ws)            // [N, 16] f32 accumulators
{
    __shared__ float sacc[16 * 16];

    const int b     = blockIdx.x / SPLITS;
    const int chunk = blockIdx.x % SPLITS;
    const int tid   = threadIdx.x;
    const int wave  = tid >> 5;
    const int lane  = tid & 31;
    const int half  = lane >> 4;   // 0: K 0/1 side, 1: K 2/3 side
    const int row   = lane & 15;   // M (for A) / N-channel (for B)

    sacc[tid] = 0.0f;
    __syncthreads();

    const int waveBase = chunk * PIX_PER_BLOCK + wave * PIX_PER_WAVE;

    // A: this lane's instance-mask row, offset to its wave chunk + half
    const float* __restrict__ aptr =
        masks + (size_t)(b * IPI + row) * HW + waveBase + half * 2;

    // B: this lane's channel field (dummy pointer for ch >= 8, fixed up below)
    const float* __restrict__ fbase;
    if      (row < 4) fbase = quat   + (size_t)(b * 4 + row)       * HW;
    else if (row < 7) fbase = scales + (size_t)(b * 3 + (row - 4)) * HW;
    else              fbase = z      + (size_t)b * HW;   // ch 7 real, ch>=8 dummy
    const float* __restrict__ fptr = fbase + waveBase + half * 2;

    // branch-free channel fixup: bb = raw*fm + fa  (ch<8: raw, ch8: 1, ch>8: 0)
    const float fm = (row < 8)  ? 1.0f : 0.0f;
    const float fa = (row == 8) ? 1.0f : 0.0f;

    v8f c = {0.f, 0.f, 0.f, 0.f, 0.f, 0.f, 0.f, 0.f};

    for (int s = 0; s < STEPS; ++s) {
        const int po = s * 4;
        __builtin_prefetch(aptr + po + 512, 0, 1);     // global_prefetch_b8

        v2f a   = *(const v2f*)(aptr + po);            // global_load_b64
        v2f raw = *(const v2f*)(fptr + po);            // global_load_b64
        v2f bb;
        bb.x = __builtin_fmaf(raw.x, fm, fa);          // two v_fma_f32, no EXEC churn
        bb.y = __builtin_fmaf(raw.y, fm, fa);

        // D = A(16x4) * B(4x16) + C   -- exact f32 accumulation
        c = __builtin_amdgcn_wmma_f32_16x16x4_f32(
                /*neg_a=*/false, a, /*neg_b=*/false, bb,
                /*c_mod=*/(short)0, c, /*reuse_a=*/false, /*reuse_b=*/false);
    }

    // cross-wave combine in LDS (ds_add_f32)
    union { v8f v; float f[8]; } u; u.v = c;
#pragma unroll
    for (int v = 0; v < 8; ++v)
        atomicAdd(&sacc[(v + 8 * half) * 16 + row], u.f[v]);
    __syncthreads();

    // one global f32 atomic per (instance, used-channel) per block
    const int m = tid >> 4, ncol = tid & 15;
    if (ncol < NCH)
        atomicAdd(&ws[(size_t)(b * IPI + m) * 16 + ncol], sacc[tid]);
}

// ---------------- kernel 2: streaming pass + fused class-id max ----------------
// out.masks = masks ; out.xy_masked[n,c] = masks[n] * xy[b,c]
// class candidate: max over pixels of (mask>0 ? cat_mask : 0)
// Exactly BLOCKS_PER_INST blocks per instance -> blockIdx maps to a single n.
__global__ __launch_bounds__(256) void mask_xy_class_kernel(
    const float* __restrict__ masks, const float* __restrict__ xy,
    const int* __restrict__ cat, float* __restrict__ out,
    int* __restrict__ wsi)
{
    __shared__ int red[256];
    const int n   = blockIdx.x / BLOCKS_PER_INST;           // instance
    const int blk = blockIdx.x % BLOCKS_PER_INST;
    const int tid = threadIdx.x;
    const int b   = n >> 4;
    const int p   = (blk * 256 + tid) * 4;                  // pixel base (quad)

    v4f m4 = *(const v4f*)(masks + (size_t)n * HW + p);           // global_load_b128
    *(v4f*)(out + OFF_MASKS + (size_t)n * HW + p) = m4;           // global_store_b128

    v4f x0 = *(const v4f*)(xy + ((size_t)b * 2 + 0) * HW + p);
    v4f x1 = *(const v4f*)(xy + ((size_t)b * 2 + 1) * HW + p);
    *(v4f*)(out + OFF_XY + ((size_t)n * 2 + 0) * HW + p) = m4 * x0;
    *(v4f*)(out + OFF_XY + ((size_t)n * 2 + 1) * HW + p) = m4 * x1;

    // fused class-id reduction (binary mask -> select then max)
    v4i c4 = *(const v4i*)(cat + (size_t)b * HW + p);
    int mx = 0;
    mx = max(mx, m4.x > 0.0f ? c4.x : 0);
    mx = max(mx, m4.y > 0.0f ? c4.y : 0);
    mx = max(mx, m4.z > 0.0f ? c4.z : 0);
    mx = max(mx, m4.w > 0.0f ? c4.w : 0);
    red[tid] = mx;
    __syncthreads();
#pragma unroll
    for (int s = 128; s > 0; s >>= 1) {
        if (tid < s) red[tid] = max(red[tid], red[tid + s]);
        __syncthreads();
    }
    if (tid == 0) atomicMax(&wsi[n], red[0]);               // 1 atomic per block
}

// ---------------- kernel 3: finalize per-instance scalars ----------------
__global__ void finalize_kernel(const float* __restrict__ ws,
                                const int* __restrict__ wsi,
                                float* __restrict__ out) {
    const int n = threadIdx.x;
    if (n >= NN) return;
    const float* s = ws + (size_t)n * 16;
    const float ms = fmaxf(s[8], 1.0f);          // mask_size (never 0 in practice)

    float q0 = s[0] / ms, q1 = s[1] / ms, q2 = s[2] / ms, q3 = s[3] / ms;
    float nrm = fmaxf(sqrtf(q0 * q0 + q1 * q1 + q2 * q2 + q3 * q3), 1e-12f);
    out[OFF_QUAT + (size_t)n * 4 + 0] = q0 / nrm;
    out[OFF_QUAT + (size_t)n * 4 + 1] = q1 / nrm;
    out[OFF_QUAT + (size_t)n * 4 + 2] = q2 / nrm;
    out[OFF_QUAT + (size_t)n * 4 + 3] = q3 / nrm;

    out[OFF_SC + (size_t)n * 3 + 0] = s[4] / ms;
    out[OFF_SC + (size_t)n * 3 + 1] = s[5] / ms;
    out[OFF_SC + (size_t)n * 3 + 2] = s[6] / ms;

    out[OFF_Z + n]     = expf(s[7] / ms);
    out[OFF_SIDS + n]  = (float)(n >> 4);        // sample_ids = n / INST_PER_IMG
    out[OFF_CLASS + n] = (float)wsi[n];
}

// ---------------- launcher ----------------
extern "C" void kernel_launch(void* const* d_in, const int* in_sizes, int n_in,
                              void* d_out, int out_size, void* d_ws, size_t ws_size,
                              hipStream_t stream) {
    const int*   cat    = (const int*)  d_in[0];  // cat_mask [B,H,W] int32
    const float* masks  = (const float*)d_in[1];  // instance_masks [N,H,W] f32
    const float* quat   = (const float*)d_in[3];  // quaternion [B,4,H,W] f32
    const float* scales = (const float*)d_in[4];  // scales [B,3,H,W] f32
    const float* xy     = (const float*)d_in[5];  // xy [B,2,H,W] f32
    const float* z      = (const float*)d_in[6];  // z [B,H,W] f32
    float* out = (float*)d_out;
    float* ws  = (float*)d_ws;                    // f32 sums
    int*   wsi = (int*)d_ws + NN * 16;            // i32 class max

    zero_ws_kernel<<<(WS_WORDS + 255) / 256, 256, 0, stream>>>((int*)d_ws, WS_WORDS);
    inst_sums_wmma_kernel<<<Bsz * SPLITS, 256, 0, stream>>>(masks, quat, scales, z, ws);
    mask_xy_class_kernel<<<NN * BLOCKS_PER_INST, 256, 0, stream>>>(masks, xy, cat, out, wsi);
    finalize_kernel<<<1, 128, 0, stream>>>(ws, wsi, out);
}